// MLPKANLayer_9131100472042
// MI455X (gfx1250) — compile-verified
//
#include <hip/hip_runtime.h>
#include <cstdint>

#define IN   64
#define OUT  64
#define BT   32   // batches per block
#define BPT  8    // batches per thread (256 thr = 64 o * 4 subtiles)

typedef unsigned int v4u __attribute__((ext_vector_type(4)));
typedef int          v8i __attribute__((ext_vector_type(8)));
typedef int          v4i __attribute__((ext_vector_type(4)));

__global__ __launch_bounds__(256) void kan_tiny_mlp_kernel(
    const float* __restrict__ x,
    const float* __restrict__ w1, const float* __restrict__ b1,
    const float* __restrict__ w2, const float* __restrict__ b2,
    const float* __restrict__ w3, const float* __restrict__ b3,
    float* __restrict__ out)
{
    __shared__ float xs[BT][IN];   // 8 KB x tile

    const int tid = threadIdx.x;
    const int o   = tid & (OUT - 1);
    const int sub = tid >> 6;
    const int b0  = blockIdx.x * BT;

    // ---------- Stage x tile (contiguous 8 KB) into LDS via TDM ----------
#if __has_builtin(__builtin_amdgcn_tensor_load_to_lds)
    {
        const uint64_t ga   = (uint64_t)(uintptr_t)(x + (size_t)b0 * IN);
        const uint32_t lds  = (uint32_t)(uintptr_t)(&xs[0][0]); // LDS byte offset
        const uint32_t nel  = BT * IN;                          // 2048 x 4B elements

        // D# group 0: count=1 (valid, user), lds_addr, 57b global_addr, type=2
        v4u g0;
        g0[0] = 1u;
        g0[1] = lds;
        g0[2] = (uint32_t)(ga & 0xffffffffu);
        g0[3] = (uint32_t)((ga >> 32) & 0x01ffffffu) | (2u << 30);

        // D# group 1: mask=0 (not in cluster), data_size=2 (4B),
        // tensor_dim0 = tile_dim0 = 2048, tensor_dim1 = tile_dim1 = 1,
        // tensor_dim0_stride = 2048.
        v8i g1;
        g1[0] = (int)(2u << 16);                        // data_size=4B
        g1[1] = (int)((nel & 0xffffu) << 16);           // tensor_dim0[15:0]
        g1[2] = (int)((nel >> 16) | (1u << 16));        // tensor_dim0[31:16], tensor_dim1=1
        g1[3] = (int)((nel & 0xffffu) << 16);           // tile_dim0
        g1[4] = 1;                                      // tile_dim1=1, tile_dim2=0
        g1[5] = (int)nel;                               // tensor_dim0_stride[31:0]
        g1[6] = 0;
        g1[7] = 0;
        const v4i gz4 = {0, 0, 0, 0};
        const v8i gz8 = {0, 0, 0, 0, 0, 0, 0, 0};

        if (tid < 32) {                                 // one wave issues the DMA
            __builtin_amdgcn_tensor_load_to_lds(g0, g1, gz4, gz4, gz8, 0);
            __builtin_amdgcn_s_wait_tensorcnt(0);
        }
    }
#else
    {
        const float*  gsrc = x + (size_t)b0 * IN;
        const uint32_t lds = (uint32_t)(uintptr_t)(&xs[0][0]);
        #pragma unroll
        for (int c = 0; c < 2; ++c) {                   // 512 x 16B chunks / 256 thr
            const int idx = tid + c * 256;
            const uint32_t loff = lds + (uint32_t)idx * 16u;
            const float* gp = gsrc + idx * 4;
            asm volatile("global_load_async_to_lds_b128 %0, %1, off"
                         :: "v"(loff), "v"(gp) : "memory");
        }
        asm volatile("s_wait_asynccnt 0" ::: "memory");
    }
#endif
    __syncthreads();

    // ---------- Main FP32 VALU pipeline ----------
    float acc[BPT];
    #pragma unroll
    for (int r = 0; r < BPT; ++r) acc[r] = 0.0f;

    const float2* __restrict__ W1p = (const float2*)w1;
    const float2* __restrict__ B1p = (const float2*)b1;
    const float4* __restrict__ W2p = (const float4*)w2;
    const float2* __restrict__ B2p = (const float2*)b2;
    const float2* __restrict__ W3p = (const float2*)w3;

    #pragma unroll 2
    for (int i = 0; i < IN; ++i) {
        const int io = i * OUT + o;
        const float2 W1 = W1p[io];
        const float2 B1 = B1p[io];
        const float4 W2 = W2p[io];     // [k0h0, k0h1, k1h0, k1h1]
        const float2 B2 = B2p[io];
        const float2 W3 = W3p[io];
        const float  B3 = b3[io];

        if (i + 4 < IN)                 // keep the L2-hot param stream ahead
            __builtin_prefetch(&W2p[(i + 4) * OUT + o], 0, 3);

        #pragma unroll
        for (int r = 0; r < BPT; ++r) {
            const float xv = xs[sub * BPT + r][i];      // LDS broadcast per wave
            float t0 = fmaxf(fmaf(xv, W1.x, B1.x), 0.0f);
            float t1 = fmaxf(fmaf(xv, W1.y, B1.y), 0.0f);
            float u0 = fmaxf(fmaf(t1, W2.y, fmaf(t0, W2.x, B2.x)), 0.0f);
            float u1 = fmaxf(fmaf(t1, W2.w, fmaf(t0, W2.z, B2.y)), 0.0f);
            acc[r] += fmaf(u1, W3.y, fmaf(u0, W3.x, B3));   // b3 folded as seed
        }
    }

    // ---------- Coalesced store ----------
    float* op = out + ((size_t)(b0 + sub * BPT)) * OUT + o;
    #pragma unroll
    for (int r = 0; r < BPT; ++r)
        op[r * OUT] = acc[r];
}

extern "C" void kernel_launch(void* const* d_in, const int* in_sizes, int n_in,
                              void* d_out, int out_size, void* d_ws, size_t ws_size,
                              hipStream_t stream) {
    const float* x  = (const float*)d_in[0];
    const float* w1 = (const float*)d_in[1];
    const float* b1 = (const float*)d_in[2];
    const float* w2 = (const float*)d_in[3];
    const float* b2 = (const float*)d_in[4];
    const float* w3 = (const float*)d_in[5];
    const float* b3 = (const float*)d_in[6];
    float* out = (float*)d_out;
    (void)in_sizes; (void)n_in; (void)out_size; (void)d_ws; (void)ws_size;

    dim3 grid(8192 / BT);   // 256 blocks
    dim3 block(256);        // 8 waves (wave32)
    hipLaunchKernelGGL(kan_tiny_mlp_kernel, grid, block, 0, stream,
                       x, w1, b1, w2, b2, w3, b3, out);
}